// Multi_Class_Affinity_Propagate_68281390072045
// MI455X (gfx1250) — compile-verified
//
#include <hip/hip_runtime.h>
#include <hip/hip_bf16.h>
#include <stdint.h>

#define NCLS    21
#define HH      256
#define WW      256
#define NPLANE  (4 * NCLS)          // 84 (b,c) planes
#define TW      64
#define TH      16
#define LSTRIDE 72                  // padded LDS row: halo-L @3, interior @4..67, halo-R @68
#define LROWS   (TH + 2)            // 18
#define LDS_FLOATS (LROWS * LSTRIDE)

// Low 32 bits of a generic pointer into __shared__ == workgroup-relative LDS
// byte offset (AMDGPU AS0->AS3 lowering is a 64->32 truncation).
__device__ __forceinline__ unsigned lds_byte_off(const void* p) {
    return (unsigned)(size_t)p;
}

__device__ __forceinline__ void async_b128(unsigned lds, const void* g) {
    asm volatile("global_load_async_to_lds_b128 %0, %1, off"
                 :: "v"(lds), "v"((uint64_t)g) : "memory");
}
__device__ __forceinline__ void async_b32(unsigned lds, const void* g) {
    asm volatile("global_load_async_to_lds_b32 %0, %1, off"
                 :: "v"(lds), "v"((uint64_t)g) : "memory");
}

// PADS order k=0..7 -> neighbor offsets (di,dj)
__constant__ const int kDI[8] = { 1, 1, 1, 0, 0, -1, -1, -1 };
__constant__ const int kDJ[8] = { 1, 0,-1, 1,-1,  1,  0, -1 };

template <bool INTERIOR>
__device__ __forceinline__ void compute_tile(const float* __restrict__ gb,
                                             const float* __restrict__ bp,
                                             float* __restrict__ op,
                                             const float* smem,
                                             int x0, int y0, int tx, int tq, int lx)
{
    #pragma unroll
    for (int r = 0; r < 4; ++r) {
        const int yy = (r << 2) + tq;       // 0..15 within tile
        const int y  = y0 + yy;
        const int x  = x0 + tx;
        const int ly = yy + 1;

        float gk[8], mk[8];
        float A = 0.0f;
        #pragma unroll
        for (int k = 0; k < 8; ++k) {
            const int ny = y + kDI[k];
            const int nx = x + kDJ[k];
            float g;
            if (INTERIOR) {
                // guidance streams (each element used once/iter): non-temporal,
                // keep the L2-resident mask/blur working set untouched
                g = __builtin_nontemporal_load(
                        gb + (((size_t)k * NCLS) << 16) + (ny << 8) + nx);
            } else {
                const bool ok = ((unsigned)ny < (unsigned)HH) &
                                ((unsigned)nx < (unsigned)WW);
                g = 0.0f;
                if (ok)
                    g = __builtin_nontemporal_load(
                            gb + (((size_t)k * NCLS) << 16) + (ny << 8) + nx);
            }
            gk[k] = g;
            A += __builtin_fabsf(g);
            mk[k] = smem[(ly + kDI[k]) * LSTRIDE + (lx + kDJ[k])];
        }

        const float invA = __builtin_amdgcn_rcpf(A);
        float gs = 0.0f, acc = 0.0f;
        #pragma unroll
        for (int k = 0; k < 8; ++k) {
            const float w = gk[k] * invA;   // normalized gate
            gs  += w;
            acc += w * mk[k];
        }

        const int idx = (y << 8) + x;
        op[idx] = (1.0f - gs) * bp[idx] + acc;
    }
}

__global__ __launch_bounds__(256)
void affprop_step(const float* __restrict__ guid,   // (4, 8*21, 256, 256)
                  const float* __restrict__ blur,   // (4, 21, 256, 256)
                  const float* __restrict__ m_in,   // (4, 21, 256, 256)
                  float* __restrict__ m_out)        // (4, 21, 256, 256)
{
    __shared__ float smem[LDS_FLOATS];

    const int p  = blockIdx.z;            // plane = b*21 + c
    const int b  = p / NCLS;
    const int c  = p - b * NCLS;
    const int x0 = blockIdx.x * TW;
    const int y0 = blockIdx.y * TH;

    const size_t plane = (size_t)p << 16; // 256*256
    const float* mp = m_in + plane;
    const float* bp = blur + plane;
    float*       op = m_out + plane;
    // guidance plane for (b, k, c) lives at ((b*168 + k*21 + c) << 16)
    const float* gb = guid + (((size_t)b * (8 * NCLS) + c) << 16);

    const int tid = threadIdx.x;

    // ---- Phase A: interior tile columns, 18 rows x 16 float4 chunks,
    // ---- staged with async b128 copies (16B aligned on both sides).
    for (int e = tid; e < LROWS * 16; e += 256) {
        const int row = e >> 4;
        const int q   = e & 15;
        const int gy  = y0 + row - 1;
        const int li  = row * LSTRIDE + 4 + (q << 2);   // byte off = row*288+16+16q
        if ((unsigned)gy < (unsigned)HH) {
            async_b128(lds_byte_off(&smem[li]), mp + (gy << 8) + x0 + (q << 2));
        } else {
            *(float4*)&smem[li] = make_float4(0.f, 0.f, 0.f, 0.f);
        }
    }
    // ---- Phase B: halo columns (x0-1 and x0+TW), 18 rows x 2 sides.
    if (tid < LROWS * 2) {
        const int row  = tid >> 1;
        const int side = tid & 1;
        const int gy   = y0 + row - 1;
        const int gx   = side ? (x0 + TW) : (x0 - 1);
        const int li   = row * LSTRIDE + (side ? (4 + TW) : 3);
        if (((unsigned)gy < (unsigned)HH) & ((unsigned)gx < (unsigned)WW)) {
            async_b32(lds_byte_off(&smem[li]), mp + (gy << 8) + gx);
        } else {
            smem[li] = 0.0f;   // disjoint from async destinations: no ordering hazard
        }
    }
    asm volatile("s_wait_asynccnt 0" ::: "memory"); // wave waits for its own copies
    __syncthreads();

    const int tx = tid & (TW - 1);   // 0..63
    const int tq = tid >> 6;         // 0..3
    const int lx = tx + 4;           // interior starts at LDS col 4

    const bool interior = (x0 != 0) & (x0 + TW != WW) & (y0 != 0) & (y0 + TH != HH);
    if (interior) {
        compute_tile<true >(gb, bp, op, smem, x0, y0, tx, tq, lx);
    } else {
        compute_tile<false>(gb, bp, op, smem, x0, y0, tx, tq, lx);
    }
}

extern "C" void kernel_launch(void* const* d_in, const int* in_sizes, int n_in,
                              void* d_out, int out_size, void* d_ws, size_t ws_size,
                              hipStream_t stream) {
    const float* guid = (const float*)d_in[0];   // (4,168,256,256) f32
    const float* blur = (const float*)d_in[1];   // (4,21,256,256)  f32
    float* out = (float*)d_out;                  // (4,21,256,256)  f32
    float* tmp = (float*)d_ws;                   // needs 84*65536*4 = 22 MB

    const dim3 grid(WW / TW, HH / TH, NPLANE);   // 4 x 16 x 84
    const dim3 block(256);

    const float* cur = blur;                     // iteration 0 reads blur_mask
    for (int t = 0; t < 24; ++t) {
        float* dst = (t & 1) ? out : tmp;        // t=23 (last) -> d_out
        affprop_step<<<grid, block, 0, stream>>>(guid, blur, cur, dst);
        cur = dst;
    }
}